// SdkBenchmark7ptStencilSpmvModel_3083786518603
// MI455X (gfx1250) — compile-verified
//
#include <hip/hip_runtime.h>
#include <stdint.h>

// ---------------------------------------------------------------------------
// 7-point variable-coefficient stencil on (B,H,W,Z)=(2,256,256,256) f32
// + deterministic MSE / max-abs-error reduction.
// Memory-bound: ~406 MB @ 23.3 TB/s => ~17us. Strategy:
//   - async global->LDS staging of 4 consecutive-w center z-rows (ASYNCcnt)
//   - z+-1 halo AND interior west/east neighbors served from LDS
//     (only w-halo rows and h+-1 rows hit global; those stay hot in 192MB L2)
//   - non-temporal hints on the pure streams (ref loads, output stores)
// ---------------------------------------------------------------------------

typedef float v4f __attribute__((ext_vector_type(4)));
typedef int   v4i __attribute__((vector_size(16)));    // matches builtin proto
typedef __attribute__((address_space(3))) float lds_f;
typedef __attribute__((address_space(1))) v4i g_v4i;   // AS1 ("__device__") int4
typedef __attribute__((address_space(3))) v4i l_v4i;   // AS3 ("__shared__") int4

#define BATCH 2
#define HDIM 256
#define WDIM 256
#define ZDIM 256
#define NROWS (BATCH * HDIM * WDIM)           /* 131072 rows of 256 z */
#define NELEM ((size_t)NROWS * (size_t)ZDIM)  /* 33554432 */
#define ROWS_PER_BLOCK 4
#define NBLOCKS (NROWS / ROWS_PER_BLOCK)      /* 32768 */
#define Z4 (ZDIM / 4)                         /* 64 float4 per row */

__device__ __forceinline__ void async_row_copy_b128(const float* gsrc, lds_f* ldst) {
#if __has_builtin(__builtin_amdgcn_global_load_async_to_lds_b128)
    __builtin_amdgcn_global_load_async_to_lds_b128(
        (g_v4i*)const_cast<float*>(gsrc), (l_v4i*)ldst, 0, 0);
#else
    const unsigned lds_byte_addr = (unsigned)(uintptr_t)ldst;
    asm volatile("global_load_async_to_lds_b128 %0, %1, off"
                 :
                 : "v"(lds_byte_addr), "v"(gsrc)
                 : "memory");
#endif
}

__device__ __forceinline__ void wait_asynccnt0() {
#if __has_builtin(__builtin_amdgcn_s_wait_asynccnt)
    __builtin_amdgcn_s_wait_asynccnt(0);
#else
    asm volatile("s_wait_asynccnt 0x0" ::: "memory");
#endif
}

__global__ __launch_bounds__(256) void stencil7_kernel(
    const float* __restrict__ x, const float* __restrict__ coeff,
    const float* __restrict__ ref, float* __restrict__ out,
    double* __restrict__ psum, float* __restrict__ pmax)
{
    __shared__ __align__(16) float rowbuf[ROWS_PER_BLOCK * ZDIM]; // 4 KB: 4 center rows
    __shared__ double ssum[256];
    __shared__ float  smax[256];

    const int tid = threadIdx.x;
    const int r   = tid >> 6;      // row within block: 0..3 (wave-uniform)
    const int z4  = tid & 63;      // float4 index along z
    const int z   = z4 << 2;
    const int R   = blockIdx.x * ROWS_PER_BLOCK + r;  // global row id
    const int w   = R & (WDIM - 1);                   // w = w0 + r, w0 4-aligned
    const int h   = (R >> 8) & (HDIM - 1);            // same h for whole block

    // ---- stage the block's 4 center rows into LDS (1 x async b128 per lane) ----
    const float* gsrc = x + (size_t)R * ZDIM + z;
    lds_f* ldst = (lds_f*)rowbuf + (size_t)tid * 4;   // tid*16 bytes into rowbuf
    async_row_copy_b128(gsrc, ldst);

    // ---- neighbors that must come from global: h+-1 rows, and w-halo rows ----
    const v4f zero4 = {0.f, 0.f, 0.f, 0.f};
    const v4f* xv = (const v4f*)x;
    const size_t q = (size_t)R * Z4 + z4;
    v4f xs = (h > 0)        ? xv[q - (size_t)WDIM * Z4] : zero4;   // south (h-1)
    v4f xn = (h < HDIM - 1) ? xv[q + (size_t)WDIM * Z4] : zero4;   // north (h+1)
    v4f xw_halo = zero4, xe_halo = zero4;
    if (r == 0 && w > 0)                xw_halo = xv[q - Z4];      // row w0-1
    if (r == ROWS_PER_BLOCK - 1 && w < WDIM - 1) xe_halo = xv[q + Z4]; // row w0+4

    // per-(b,h,w) coefficients: 28 contiguous bytes shared by all 64 lanes of the row
    const float* cp = coeff + (size_t)R * 7;
    const float c0 = cp[0], c1 = cp[1], c2 = cp[2], c3 = cp[3];
    const float c4 = cp[4], c5 = cp[5], c6 = cp[6];

    wait_asynccnt0();
    __syncthreads();

    // ---- center + z-halo from LDS ----
    const float* rb = &rowbuf[r * ZDIM];
    v4f xc = *(const v4f*)&rb[z];
    const float bl = (z4 == 0)  ? 0.f : rb[z - 1];
    const float tr = (z4 == 63) ? 0.f : rb[z + 4];
    v4f bot = {bl, xc.x, xc.y, xc.z};   // x[z-1]
    v4f top = {xc.y, xc.z, xc.w, tr};   // x[z+1]

    // ---- west/east: interior rows read the LDS copy staged by sibling waves ----
    v4f xw = (r > 0)                  ? *(const v4f*)&rowbuf[(r - 1) * ZDIM + z] : xw_halo;
    v4f xe = (r < ROWS_PER_BLOCK - 1) ? *(const v4f*)&rowbuf[(r + 1) * ZDIM + z] : xe_halo;

    v4f o = c0 * xw + c1 * xe + c2 * xs + c3 * xn + c4 * bot + c5 * top + c6 * xc;

    // ---- streaming output (NT so it doesn't evict x from L2) ----
    float* op = out + 1 + (size_t)R * ZDIM + z;   // +1: d_out[0] is the loss slot
    __builtin_nontemporal_store(o.x, op + 0);
    __builtin_nontemporal_store(o.y, op + 1);
    __builtin_nontemporal_store(o.z, op + 2);
    __builtin_nontemporal_store(o.w, op + 3);

    // ---- error reduction (deterministic) ----
    v4f rv = __builtin_nontemporal_load(&((const v4f*)ref)[q]);
    v4f e = o - rv;
    double ds = (double)e.x * e.x + (double)e.y * e.y +
                (double)e.z * e.z + (double)e.w * e.w;
    float m = fmaxf(fmaxf(fabsf(e.x), fabsf(e.y)), fmaxf(fabsf(e.z), fabsf(e.w)));

    ssum[tid] = ds;
    smax[tid] = m;
    __syncthreads();
#pragma unroll
    for (int s = 128; s > 0; s >>= 1) {
        if (tid < s) {
            ssum[tid] += ssum[tid + s];
            smax[tid] = fmaxf(smax[tid], smax[tid + s]);
        }
        __syncthreads();
    }
    if (tid == 0) {
        psum[blockIdx.x] = ssum[0];
        pmax[blockIdx.x] = smax[0];
    }
}

__global__ __launch_bounds__(256) void finalize_kernel(
    const double* __restrict__ psum, const float* __restrict__ pmax,
    float* __restrict__ out)
{
    __shared__ double ssum[256];
    __shared__ float  smax[256];
    const int tid = threadIdx.x;
    double s = 0.0;
    float  m = 0.0f;
    for (int i = tid; i < NBLOCKS; i += 256) {   // fixed order per thread
        s += psum[i];
        m = fmaxf(m, pmax[i]);
    }
    ssum[tid] = s;
    smax[tid] = m;
    __syncthreads();
#pragma unroll
    for (int sft = 128; sft > 0; sft >>= 1) {
        if (tid < sft) {
            ssum[tid] += ssum[tid + sft];
            smax[tid] = fmaxf(smax[tid], smax[tid + sft]);
        }
        __syncthreads();
    }
    if (tid == 0) {
        out[0]         = (float)(ssum[0] / (double)NELEM);  // loss
        out[NELEM + 1] = smax[0];                           // max_abs_error
    }
}

extern "C" void kernel_launch(void* const* d_in, const int* in_sizes, int n_in,
                              void* d_out, int out_size, void* d_ws, size_t ws_size,
                              hipStream_t stream) {
    const float* x     = (const float*)d_in[0];
    const float* coeff = (const float*)d_in[1];
    const float* ref   = (const float*)d_in[2];
    float* out = (float*)d_out;

    double* psum = (double*)d_ws;                                    // 32768 * 8B
    float*  pmax = (float*)((char*)d_ws + NBLOCKS * sizeof(double)); // + 32768 * 4B

    stencil7_kernel<<<NBLOCKS, 256, 0, stream>>>(x, coeff, ref, out, psum, pmax);
    finalize_kernel<<<1, 256, 0, stream>>>(psum, pmax, out);
}